// TaxaNetLoss_8942121910892
// MI455X (gfx1250) — compile-verified
//
#include <hip/hip_runtime.h>
#include <math.h>

// Max row length staged in LDS (N = 5000 for this problem; 5120*4 = 20 KB/block,
// so 16 workgroups per WGP fit in the 320 KB LDS pool).
#define LDS_FLOATS 5120

// ---------------------------------------------------------------------------
// Kernel A: one wave32 per (b, l) row of y_pred.
//   - Async-DMA the row from global into LDS in 512B b128 chunks (ASYNCcnt),
//     non-temporal (328 MB streamed once > 192 MB L2 -> don't pollute L2).
//   - Pass 1 (from LDS): max + first-occurrence argmax.
//   - Pass 2 (from LDS): sum of exp(x - max).
//   - Wave32 butterfly reductions; lane 0 writes argmax + CE term to workspace.
// ---------------------------------------------------------------------------
__global__ __launch_bounds__(32)
void row_softmax_stats(const float* __restrict__ y_pred,
                       const int*   __restrict__ y_true,
                       int N,
                       int*   __restrict__ idx_out,
                       float* __restrict__ ce_out)
{
    __shared__ __attribute__((aligned(16))) float smem[LDS_FLOATS];

    const int row  = blockIdx.x;       // row = b*L + l
    const int lane = threadIdx.x;      // wave32: 0..31
    const float* g = y_pred + (size_t)row * (size_t)N;

    const int full       = N >> 7;     // #chunks of 128 floats (512 B)
    const int tail_start = full << 7;

    // ---- Issue async DMA: global -> LDS, one b128 per lane per chunk ------
    // LDS byte address = low 32 bits of the generic shared-memory address.
    // INST_OFFSET is added to BOTH the LDS and global addresses, so 4 chunks
    // share one address-register pair via immediate offsets.
    unsigned vdst = (unsigned)(unsigned long long)(&smem[0]) + (unsigned)lane * 16u;
    unsigned voff = (unsigned)((size_t)row * (size_t)N * 4u) + (unsigned)lane * 16u;
    int k = 0;
    for (; k + 4 <= full; k += 4) {
        asm volatile(
            "global_load_async_to_lds_b128 %0, %1, %2 offset:0 th:TH_LOAD_NT\n\t"
            "global_load_async_to_lds_b128 %0, %1, %2 offset:512 th:TH_LOAD_NT\n\t"
            "global_load_async_to_lds_b128 %0, %1, %2 offset:1024 th:TH_LOAD_NT\n\t"
            "global_load_async_to_lds_b128 %0, %1, %2 offset:1536 th:TH_LOAD_NT"
            :: "v"(vdst), "v"(voff), "s"(y_pred)
            : "memory");
        vdst += 2048u;
        voff += 2048u;
    }
    for (; k < full; ++k) {
        asm volatile(
            "global_load_async_to_lds_b128 %0, %1, %2 offset:0 th:TH_LOAD_NT"
            :: "v"(vdst), "v"(voff), "s"(y_pred)
            : "memory");
        vdst += 512u;
        voff += 512u;
    }

    // ---- Tail (< 128 floats) via regular NT loads, kept in registers ------
    float tx[4];
    int   tn = 0;
    for (int t = tail_start + lane; t < N; t += 32)
        tx[tn++] = __builtin_nontemporal_load(&g[t]);

    // Target logit (uniform gather)
    const float xt = __builtin_nontemporal_load(&g[y_true[row]]);

    // Wait for the DMA engine to land all chunks in LDS (single wave owns it;
    // no barrier needed, asm memory clobber orders the LDS reads below).
    asm volatile("s_wait_asynccnt 0x0" ::: "memory");

    // ---- Pass 1: max + argmax (first occurrence, to match jnp.argmax) -----
    float m  = -INFINITY;
    int   am = 0x7fffffff;
    for (int c = 0; c < full; ++c) {
        const int   base = (c << 7) + (lane << 2);
        const float4 v = *reinterpret_cast<const float4*>(&smem[base]);
        if (v.x > m) { m = v.x; am = base;     }
        if (v.y > m) { m = v.y; am = base + 1; }
        if (v.z > m) { m = v.z; am = base + 2; }
        if (v.w > m) { m = v.w; am = base + 3; }
    }
    for (int i = 0; i < tn; ++i) {
        const float v = tx[i];
        if (v > m) { m = v; am = tail_start + lane + (i << 5); }
    }
    // Wave32 butterfly: lexicographic (value desc, index asc) -> all lanes agree
    for (int off = 16; off; off >>= 1) {
        const float om = __shfl_xor(m,  off, 32);
        const int   oa = __shfl_xor(am, off, 32);
        if (om > m || (om == m && oa < am)) { m = om; am = oa; }
    }

    // ---- Pass 2: sum of exp(x - M) from LDS -------------------------------
    float s = 0.0f;
    for (int c = 0; c < full; ++c) {
        const int   base = (c << 7) + (lane << 2);
        const float4 v = *reinterpret_cast<const float4*>(&smem[base]);
        s += __expf(v.x - m) + __expf(v.y - m) + __expf(v.z - m) + __expf(v.w - m);
    }
    for (int i = 0; i < tn; ++i) s += __expf(tx[i] - m);
    for (int off = 16; off; off >>= 1) s += __shfl_xor(s, off, 32);

    if (lane == 0) {
        idx_out[row] = am;
        // -log_softmax at target: M + log(sum) - x_t
        ce_out[row] = m + __logf(s) - xt;
    }
}

// ---------------------------------------------------------------------------
// Kernel B: single block, deterministic tree reduction.
//   loss = sum_{j=1..3} w[j] * ( count(H[idx[:,j-1], idx[:,j]] != 1) * e
//                                + mean_b ce[b, j] )
// ---------------------------------------------------------------------------
__global__ __launch_bounds__(256)
void taxanet_finalize(const int*   __restrict__ idxs,
                      const float* __restrict__ ce,
                      const float* __restrict__ H,
                      int B, int L, int N,
                      float* __restrict__ out)
{
    const int tid = threadIdx.x;
    float pen0 = 0.f, pen1 = 0.f, pen2 = 0.f;
    float ce0  = 0.f, ce1  = 0.f, ce2  = 0.f;

    for (int b = tid; b < B; b += 256) {
        const int* ib = idxs + b * L;
        const int p0 = ib[0], p1 = ib[1], p2 = ib[2], p3 = ib[3];
        pen0 += (H[(size_t)p0 * (size_t)N + p1] != 1.0f) ? 1.0f : 0.0f;
        pen1 += (H[(size_t)p1 * (size_t)N + p2] != 1.0f) ? 1.0f : 0.0f;
        pen2 += (H[(size_t)p2 * (size_t)N + p3] != 1.0f) ? 1.0f : 0.0f;
        const float* cb = ce + b * L;
        ce0 += cb[1];
        ce1 += cb[2];
        ce2 += cb[3];
    }

    __shared__ float red[6][256];
    red[0][tid] = pen0; red[1][tid] = pen1; red[2][tid] = pen2;
    red[3][tid] = ce0;  red[4][tid] = ce1;  red[5][tid] = ce2;
    __syncthreads();
    for (int off = 128; off; off >>= 1) {
        if (tid < off) {
            #pragma unroll
            for (int r = 0; r < 6; ++r) red[r][tid] += red[r][tid + off];
        }
        __syncthreads();
    }

    if (tid == 0) {
        const float Ef   = 2.71828182845904523536f;
        const float invB = 1.0f / (float)B;
        const float loss = 0.25f * (red[0][0] * Ef + red[3][0] * invB)
                         + 0.15f * (red[1][0] * Ef + red[4][0] * invB)
                         + 0.10f * (red[2][0] * Ef + red[5][0] * invB);
        out[0] = loss;
    }
}

// ---------------------------------------------------------------------------
extern "C" void kernel_launch(void* const* d_in, const int* in_sizes, int n_in,
                              void* d_out, int out_size, void* d_ws, size_t ws_size,
                              hipStream_t stream)
{
    const float* y_pred = (const float*)d_in[0];   // (B, L, N) f32
    const int*   y_true = (const int*)  d_in[1];   // (B, L)    i32
    const float* H      = (const float*)d_in[2];   // (N, N)    f32

    const int rows = in_sizes[1];          // B * L
    const int L    = 4;                    // fixed by reference (len(WEIGHTS))
    const int B    = rows / L;
    const int N    = in_sizes[0] / rows;   // 5000

    // Workspace layout: [rows] argmax (int) | [rows] ce term (float)
    int*   idx_ws = (int*)d_ws;
    float* ce_ws  = (float*)((char*)d_ws + (size_t)rows * sizeof(int));

    row_softmax_stats<<<rows, 32, 0, stream>>>(y_pred, y_true, N, idx_ws, ce_ws);
    taxanet_finalize<<<1, 256, 0, stream>>>(idx_ws, ce_ws, H, B, L, N, (float*)d_out);
}